// LRLoss_12790412607829
// MI455X (gfx1250) — compile-verified
//
#include <hip/hip_runtime.h>
#include <hip/hip_bf16.h>

// ---------------- problem constants (reference setup_inputs is fixed) -------
static constexpr int IMH  = 1024;
static constexpr int IMW  = 2048;
static constexpr int IMHW = IMH * IMW;
static constexpr int PARTN = 8192;   // per-quantity partial-sum slots (>= any grid size used)
static constexpr int NQ    = 9;      // sad_l, sad_r, lr1, lr2, ssim3_l, ssim5_l, ssim3_r, ssim5_r, smooth

typedef float v2f  __attribute__((ext_vector_type(2)));
typedef float v8f  __attribute__((ext_vector_type(8)));
typedef unsigned int u32x4 __attribute__((ext_vector_type(4)));
typedef int   i32x8 __attribute__((ext_vector_type(8)));
typedef int   i32x4 __attribute__((ext_vector_type(4)));

#if defined(__gfx1250__) && __has_builtin(__builtin_amdgcn_wmma_f32_16x16x4_f32)
#define USE_WMMA_RED 1
#endif
#if defined(__gfx1250__) && __has_builtin(__builtin_amdgcn_tensor_load_to_lds) && \
    __has_builtin(__builtin_amdgcn_s_wait_tensorcnt)
#define USE_TDM 1
#endif

// ---------------- reductions ------------------------------------------------
// Wave-level sum of 32 lane values via V_WMMA_F32_16X16X4_F32 with B = ones:
// A[m][0] = p_m (lanes 0-15), A[m][2] = p_{m+16} (lanes 16-31)  =>  row-sum
// D[m] = p_m + p_{m+16}; lanes<16 hold rows 0-7 in d[0..7], lanes>=16 rows 8-15.
__device__ __forceinline__ float wave_sum32(float v) {
#ifdef USE_WMMA_RED
    v2f a; a.x = v;    a.y = 0.0f;
    v2f b; b.x = 1.0f; b.y = 1.0f;
    v8f c = {};
    v8f d = __builtin_amdgcn_wmma_f32_16x16x4_f32(false, a, false, b, (short)0, c, false, false);
    float s = d[0] + d[1] + d[2] + d[3] + d[4] + d[5] + d[6] + d[7];
    s += __shfl_xor(s, 16, 32);
    return s;
#else
    float s = v;
    for (int off = 16; off > 0; off >>= 1) s += __shfl_xor(s, off, 32);
    return s;
#endif
}

// Deterministic 256-thread block sum (8 waves). Caller must have full EXEC.
__device__ __forceinline__ float block_sum256(float v, float* wred8) {
    float s = wave_sum32(v);
    const int tid = threadIdx.x;
    if ((tid & 31) == 0) wred8[tid >> 5] = s;
    __syncthreads();
    float t = 0.0f;
#pragma unroll
    for (int i = 0; i < 8; ++i) t += wred8[i];
    __syncthreads();   // allow wred8 reuse
    return t;
}

// ---------------- TDM 2-D tile load: global (row stride IMW) -> LDS ---------
#ifdef USE_TDM
__device__ __forceinline__ unsigned lds_byte_off(const void* p) {
    // flat LDS address = aperture_base(high 32b) | byte offset(low 32b)
    return (unsigned)(unsigned long long)p;
}
__device__ __forceinline__ void tdm_load_tile_2d(const float* gbase, unsigned lds_byte,
                                                 unsigned rem_w, unsigned rem_h,
                                                 unsigned tile_w, unsigned tile_h) {
    unsigned long long ga = (unsigned long long)(const void*)gbase;
    u32x4 g0;
    g0[0] = 1u;                                          // count=1 (valid user descriptor)
    g0[1] = lds_byte;                                    // lds_addr
    g0[2] = (unsigned)(ga & 0xFFFFFFFFu);                // global_addr[31:0]
    g0[3] = (unsigned)((ga >> 32) & 0x01FFFFFFu)         // global_addr[56:32]
          | (2u << 30);                                  // type = 2 ("image")
    i32x8 g1;
    g1[0] = (int)(2u << 16);                             // data_size = 4 bytes
    g1[1] = (int)((rem_w & 0xFFFFu) << 16);              // tensor_dim0[15:0]
    g1[2] = (int)(((rem_w >> 16) & 0xFFFFu) | ((rem_h & 0xFFFFu) << 16));
    g1[3] = (int)(((rem_h >> 16) & 0xFFFFu) | (tile_w << 16));   // tile_dim0
    g1[4] = (int)(tile_h & 0xFFFFu);                     // tile_dim1 (tile_dim2 = 0)
    g1[5] = (int)IMW;                                    // tensor_dim0_stride[31:0]
    g1[6] = 0;                                           // stride[47:32], dim1_stride[15:0]
    g1[7] = 0;
    i32x4 z4 = {0, 0, 0, 0};
    i32x8 z8 = {0, 0, 0, 0, 0, 0, 0, 0};                 // VADDR4 slot (unused / NULL)
    __builtin_amdgcn_tensor_load_to_lds(g0, g1, z4, z4, z8, 0);
}
#endif

// ---------------- helpers ---------------------------------------------------
// warp1d sampling: padded row P[j] = (1<=j<=IMW) ? row[j-1] : 0, x clipped to [0, IMW+1]
__device__ __forceinline__ float samp_row(const float* __restrict__ row, float x) {
    x = fminf(fmaxf(x, 0.0f), (float)(IMW + 1));
    float x0 = floorf(x);
    float x1 = fminf(x0 + 1.0f, (float)(IMW + 1));
    int i0 = (int)x0;
    int i1 = (int)x1;
    float pl = (i0 >= 1 && i0 <= IMW) ? row[i0 - 1] : 0.0f;
    float pr = (i1 <= IMW) ? row[i1 - 1] : 0.0f;   // i1 >= 1 always
    return (x1 - x) * pl + (x - x0) * pr;
}

// ---------------- stage 0: zero partial region ------------------------------
__global__ void zero_kernel(float* __restrict__ p, int n) {
    int i = blockIdx.x * 256 + threadIdx.x;
    if (i < n) p[i] = 0.0f;
}

// ---------------- stage 1: fused warp + SAD + gray --------------------------
// grid = IMHW/256 = 8192 blocks exactly, 256 threads
__global__ void warp_kernel(const float* __restrict__ dl, const float* __restrict__ dr,
                            const float* __restrict__ left, const float* __restrict__ right,
                            float* __restrict__ gl, float* __restrict__ gr,
                            float* __restrict__ gel, float* __restrict__ ger,
                            float* __restrict__ part) {
    __shared__ float wred[8];
    const int tid = threadIdx.x;
    const int idx = blockIdx.x * 256 + tid;
    const int h = idx >> 11;         // IMW == 2048
    const int w = idx & (IMW - 1);

    const float* lrow = left  + h * IMW;   // channel stride IMHW
    const float* rrow = right + h * IMW;
    __builtin_prefetch(lrow + IMHW + w, 0, 0);   // global_prefetch_b8
    __builtin_prefetch(rrow + IMHW + w, 0, 0);

    const float dlv = dl[idx];
    const float drv = dr[idx];
    const float xr = (float)(w + 1) + drv;   // est_right samples LEFT at +disp_right
    const float xl = (float)(w + 1) - dlv;   // est_left  samples RIGHT at -disp_left

    float er0 = samp_row(lrow,            xr);
    float er1 = samp_row(lrow + IMHW,     xr);
    float er2 = samp_row(lrow + 2 * IMHW, xr);
    float el0 = samp_row(rrow,            xl);
    float el1 = samp_row(rrow + IMHW,     xl);
    float el2 = samp_row(rrow + 2 * IMHW, xl);

    const float l0 = lrow[w], l1 = lrow[IMHW + w], l2 = lrow[2 * IMHW + w];
    const float r0 = rrow[w], r1 = rrow[IMHW + w], r2 = rrow[2 * IMHW + w];

    const float sadl = fabsf(l0 - el0) + fabsf(l1 - el1) + fabsf(l2 - el2);
    const float sadr = fabsf(r0 - er0) + fabsf(r1 - er1) + fabsf(r2 - er2);

    const float third = 1.0f / 3.0f;
    gl[idx]  = (l0 + l1 + l2) * third;
    gr[idx]  = (r0 + r1 + r2) * third;
    gel[idx] = (el0 + el1 + el2) * third;
    ger[idx] = (er0 + er1 + er2) * third;

    const float l2rv = samp_row(dl + h * IMW, xr);   // warp1d(disp_left, disp_right)
    const float r2lv = samp_row(dr + h * IMW, xl);   // warp1d(disp_right, -disp_left)
    const float lr1 = fabsf(r2lv - dlv);
    const float lr2 = fabsf(l2rv - drv);

    float vals[4] = { sadl, sadr, lr1, lr2 };
#pragma unroll
    for (int q = 0; q < 4; ++q) {
        float t = block_sum256(vals[q], wred);
        if (tid == 0) part[q * PARTN + blockIdx.x] = t;
    }
}

// ---------------- stage 2: SSIM (nested VALID box pools) --------------------
// One 16x16 output tile per block; gray tiles fetched by the Tensor Data Mover
// (global -> LDS, 2-D tile with row stride IMW), mu tile computed in LDS.
template <int WW>
__global__ void ssim_kernel(const float* __restrict__ gx, const float* __restrict__ gy,
                            float* __restrict__ part) {
    constexpr int C    = (WW - 1) / 2;
    constexpr int MT   = 16 + WW - 1;        // mu-tile edge
    constexpr int GT   = 16 + 2 * (WW - 1);  // gray-tile edge
    constexpr int Hout = IMH - 2 * (WW - 1);
    constexpr int Wout = IMW - 2 * (WW - 1);

    __shared__ float sgx[GT * GT], sgy[GT * GT];
    __shared__ float smx[MT * MT], smy[MT * MT];
    __shared__ float wred[8];

    const int tid = threadIdx.x;
    const int i0 = blockIdx.y * 16;
    const int j0 = blockIdx.x * 16;

#ifdef USE_TDM
    // Wave 0 issues both tile DMAs; OOB rows/cols (beyond remaining extent)
    // return zero and are consumed only by invalidated outputs.
    if (tid < 32) {
        const unsigned rem_w = (unsigned)(IMW - j0);
        const unsigned rem_h = (unsigned)(IMH - i0);
        tdm_load_tile_2d(gx + (size_t)i0 * IMW + j0, lds_byte_off(sgx), rem_w, rem_h, GT, GT);
        tdm_load_tile_2d(gy + (size_t)i0 * IMW + j0, lds_byte_off(sgy), rem_w, rem_h, GT, GT);
        __builtin_amdgcn_s_wait_tensorcnt(0);
    }
    __syncthreads();
#else
    for (int p = tid; p < GT * GT; p += 256) {
        int rr = p / GT, cc = p % GT;
        int gi = min(i0 + rr, IMH - 1);
        int gj = min(j0 + cc, IMW - 1);
        sgx[p] = gx[gi * IMW + gj];
        sgy[p] = gy[gi * IMW + gj];
    }
    __syncthreads();
#endif

    const float inv = 1.0f / (float)(WW * WW);
    for (int p = tid; p < MT * MT; p += 256) {
        int rr = p / MT, cc = p % MT;
        float ax = 0.0f, ay = 0.0f;
#pragma unroll
        for (int a = 0; a < WW; ++a)
#pragma unroll
            for (int b = 0; b < WW; ++b) {
                ax += sgx[(rr + a) * GT + cc + b];
                ay += sgy[(rr + a) * GT + cc + b];
            }
        smx[p] = ax * inv;
        smy[p] = ay * inv;
    }
    __syncthreads();

    const int r  = tid >> 4;
    const int cc = tid & 15;
    const int oi = i0 + r;
    const int oj = j0 + cc;

    float sx = 0.0f, sy = 0.0f, sxy = 0.0f;
#pragma unroll
    for (int a = 0; a < WW; ++a)
#pragma unroll
        for (int b = 0; b < WW; ++b) {
            float mx = smx[(r + a) * MT + cc + b];
            float my = smy[(r + a) * MT + cc + b];
            float tx = sgx[(r + a + C) * GT + cc + b + C] - mx;
            float ty = sgy[(r + a + C) * GT + cc + b + C] - my;
            sx += tx * tx; sy += ty * ty; sxy += tx * ty;
        }
    sx *= inv; sy *= inv; sxy *= inv;
    float mx = smx[(r + C) * MT + cc + C];
    float my = smy[(r + C) * MT + cc + C];
    const float C1 = 1e-4f, C2 = 9e-4f;
    float n  = (2.0f * mx * my + C1) * (2.0f * sxy + C2);
    float dd = (mx * mx + my * my + C1) * (sx + sy + C2);
    float v = fminf(fmaxf(1.0f - n / dd, 0.0f), 2.0f);
    float val = (oi < Hout && oj < Wout) ? v : 0.0f;

    float t = block_sum256(val, wred);
    if (tid == 0) part[blockIdx.y * gridDim.x + blockIdx.x] = t;
}

// ---------------- stage 3: disparity smoothness -----------------------------
__global__ void smooth_kernel(const float* __restrict__ dl, const float* __restrict__ dr,
                              const float* __restrict__ gl, const float* __restrict__ gr,
                              float* __restrict__ part) {
    __shared__ float wred[8];
    const int tid = threadIdx.x;
    const int i = blockIdx.y * 16 + (tid >> 4);
    const int j = blockIdx.x * 16 + (tid & 15);
    float val = 0.0f;
    if (i < IMH - 2 && j < IMW - 2) {
        float dlap1 = 0.0f, ilap1 = 0.0f, dlap2 = 0.0f, ilap2 = 0.0f;
#pragma unroll
        for (int a = 0; a < 3; ++a)
#pragma unroll
            for (int b = 0; b < 3; ++b) {
                float wgt = (a == 1 && b == 1) ? -8.0f : 1.0f;
                int o = (i + a) * IMW + j + b;
                dlap1 += wgt * dl[o]; ilap1 += wgt * gl[o];
                dlap2 += wgt * dr[o]; ilap2 += wgt * gr[o];
            }
        val = expf(-fabsf(ilap1)) * fabsf(dlap1) + expf(-fabsf(ilap2)) * fabsf(dlap2);
    }
    float t = block_sum256(val, wred);
    if (tid == 0) part[blockIdx.y * gridDim.x + blockIdx.x] = t;
}

// ---------------- stage 4: deterministic final reduction (WMMA) -------------
__global__ void reduce_kernel(const float* __restrict__ part, float* __restrict__ reduced) {
    __shared__ float wred[8];
    const int q = blockIdx.x;
    const float* p = part + q * PARTN;
    const int tid = threadIdx.x;
    float s = 0.0f;
    for (int k = tid; k < PARTN; k += 256) s += p[k];
    float t = block_sum256(s, wred);
    if (tid == 0) reduced[q] = t;
}

// ---------------- stage 5: combine into the 3 outputs -----------------------
__global__ void combine_kernel(const float* __restrict__ red, float* __restrict__ out) {
    if (threadIdx.x == 0 && blockIdx.x == 0) {
        float sadl = red[0] / (3.0f * (float)IMHW);
        float sadr = red[1] / (3.0f * (float)IMHW);
        float lr   = (red[2] + red[3]) / (float)IMHW;
        float n3 = (float)((IMH - 4) * (IMW - 4));
        float n5 = (float)((IMH - 8) * (IMW - 8));
        float ssiml = 0.5f * (red[4] / n3) + 0.5f * (red[5] / n5);
        float ssimr = 0.5f * (red[6] / n3) + 0.5f * (red[7] / n5);
        float rec = 0.5f * ssiml + 0.5f * sadl + 0.5f * ssimr + 0.5f * sadr;
        float smooth = red[8] / (float)((IMH - 2) * (IMW - 2));
        out[0] = rec;
        out[1] = 0.1f * smooth;
        out[2] = 0.1f * lr;
    }
}

// ---------------- launch ----------------------------------------------------
extern "C" void kernel_launch(void* const* d_in, const int* in_sizes, int n_in,
                              void* d_out, int out_size, void* d_ws, size_t ws_size,
                              hipStream_t stream) {
    (void)in_sizes; (void)n_in; (void)out_size; (void)ws_size;
    const float* dl    = (const float*)d_in[0];
    const float* dr    = (const float*)d_in[1];
    const float* left  = (const float*)d_in[2];
    const float* right = (const float*)d_in[3];
    float* out = (float*)d_out;

    float* wsf = (float*)d_ws;
    float* gl  = wsf;
    float* gr  = wsf + (size_t)IMHW;
    float* gel = wsf + 2 * (size_t)IMHW;
    float* ger = wsf + 3 * (size_t)IMHW;
    float* part = wsf + 4 * (size_t)IMHW;      // NQ * PARTN floats
    float* red  = part + (size_t)NQ * PARTN;   // NQ floats (+ pad)

    const int zn = NQ * PARTN + 16;
    zero_kernel<<<(zn + 255) / 256, 256, 0, stream>>>(part, zn);

    warp_kernel<<<IMHW / 256, 256, 0, stream>>>(dl, dr, left, right,
                                                gl, gr, gel, ger, part);

    dim3 g3((IMW - 4 + 15) / 16, (IMH - 4 + 15) / 16);
    dim3 g5((IMW - 8 + 15) / 16, (IMH - 8 + 15) / 16);
    ssim_kernel<3><<<g3, 256, 0, stream>>>(gl, gel, part + 4 * PARTN);
    ssim_kernel<5><<<g5, 256, 0, stream>>>(gl, gel, part + 5 * PARTN);
    ssim_kernel<3><<<g3, 256, 0, stream>>>(gr, ger, part + 6 * PARTN);
    ssim_kernel<5><<<g5, 256, 0, stream>>>(gr, ger, part + 7 * PARTN);

    dim3 gs((IMW - 2 + 15) / 16, (IMH - 2 + 15) / 16);
    smooth_kernel<<<gs, 256, 0, stream>>>(dl, dr, gl, gr, part + 8 * PARTN);

    reduce_kernel<<<NQ, 256, 0, stream>>>(part, red);
    combine_kernel<<<1, 64, 0, stream>>>(red, out);
}